// HierarchicalLTI2_27049704030519
// MI455X (gfx1250) — compile-verified
//
#include <hip/hip_runtime.h>

// ---------------------------------------------------------------------------
// HierarchicalLTI2 on MI455X (gfx1250, wave32, WMMA + async-LDS)
//
// x_{t+1} = A x_t + B v_t   (row form: X_{t+1} = X_t A^T + W_t,  W = V B^T)
// y_t     = sum_i C_i x_{i,t}
//
// Per layer: chunked scan (chunk=16) with matrix powers P_k=(A^T)^k; the
// per-chunk local solution is a sum of 15 shifted 16x64 @ 64x64 GEMMs via
// v_wmma_f32_16x16x4_f32.  Each block handles 4 chunks so one P_k stage
// feeds 64 WMMAs, and P_k staging is double-buffered through
// global_load_async_to_lds_b128 (ASYNCcnt) overlapped with compute.
// Chunk carries are prefix-scanned with the SAME kernel at group level
// (Q_k = P_16^k), leaving a 32-step serial spine per layer.
// ---------------------------------------------------------------------------

typedef __attribute__((ext_vector_type(2))) float v2f;
typedef __attribute__((ext_vector_type(8))) float v8f;

#define T_SEQ 8192
#define NST   64
#define NLAY  32
#define CH    16
#define NCH   512   // T_SEQ / CH
#define NG    32    // groups of chunks
#define CPB   4     // chunks per tri-scan block
#define LDSP  68    // padded LDS row stride (floats); 272B = 17*16B (B128 ok)

__device__ __forceinline__ v8f wmma_f32_16x16x4(v2f a, v2f b, v8f c) {
  // (neg_a, A, neg_b, B, c_mod, C, reuse_a, reuse_b)
  return __builtin_amdgcn_wmma_f32_16x16x4_f32(false, a, false, b, (short)0, c,
                                               false, false);
}

// Issue 8 x B128 async copies: one 64x64 f32 matrix (row-major, stride 64)
// into a padded [64][LDSP] LDS buffer.  Tracked by ASYNCcnt.
__device__ __forceinline__ void async_stage_pk(const float* __restrict__ src,
                                               void* ldsbase, int tid) {
  const unsigned lbase = (unsigned)(size_t)ldsbase;  // low 32b = LDS offset
#pragma unroll
  for (int r = 0; r < 8; ++r) {
    const int tidx = tid + 128 * r;  // 0..1023 transfers of 16B
    const int row = tidx >> 4, seg = tidx & 15;
    const unsigned loff = lbase + (unsigned)(row * (LDSP * 4) + seg * 16);
    const unsigned long long g =
        (unsigned long long)(size_t)(src + row * 64 + seg * 4);
    asm volatile("global_load_async_to_lds_b128 %0, %1, off" ::"v"(loff),
                 "v"(g)
                 : "memory");
  }
}

// ---------------------------------------------------------------------------
// P_k = (A^T)^k, k=0..16 ; Q_k = (P_16)^k, k=0..16.  One workgroup.
// ---------------------------------------------------------------------------
__global__ __launch_bounds__(256) void powers_kernel(
    const float* __restrict__ Amat,  // A_i raw [r][c] (x' = A x)
    float* __restrict__ P,           // 17*4096
    float* __restrict__ Q)           // 17*4096
{
  __shared__ float M1[NST][LDSP];
  __shared__ float Cur[NST][LDSP];
  const int tid = threadIdx.x;

  for (int idx = tid; idx < 4096; idx += 256) {
    const int r = idx >> 6, c = idx & 63;
    const float id = (r == c) ? 1.f : 0.f;
    P[idx] = id;
    Q[idx] = id;
    const float at = Amat[c * 64 + r];  // A^T
    M1[r][c] = at;
    Cur[r][c] = at;
    P[4096 + idx] = at;
  }
  __syncthreads();

  for (int k = 2; k <= 16; ++k) {
    float out[16];
    for (int q = 0; q < 16; ++q) {
      const int idx = tid + 256 * q;
      const int r = idx >> 6, c = idx & 63;
      float acc = 0.f;
      for (int m = 0; m < 64; ++m) acc += Cur[r][m] * M1[m][c];
      out[q] = acc;
    }
    __syncthreads();
    for (int q = 0; q < 16; ++q) {
      const int idx = tid + 256 * q;
      Cur[idx >> 6][idx & 63] = out[q];
      P[k * 4096 + idx] = out[q];
    }
    __syncthreads();
  }

  // Q_1 = P_16 (now in Cur)
  for (int q = 0; q < 16; ++q) {
    const int idx = tid + 256 * q;
    const float v = Cur[idx >> 6][idx & 63];
    M1[idx >> 6][idx & 63] = v;
    Q[4096 + idx] = v;
  }
  __syncthreads();
  for (int k = 2; k <= 16; ++k) {
    float out[16];
    for (int q = 0; q < 16; ++q) {
      const int idx = tid + 256 * q;
      const int r = idx >> 6, c = idx & 63;
      float acc = 0.f;
      for (int m = 0; m < 64; ++m) acc += Cur[r][m] * M1[m][c];
      out[q] = acc;
    }
    __syncthreads();
    for (int q = 0; q < 16; ++q) {
      const int idx = tid + 256 * q;
      Cur[idx >> 6][idx & 63] = out[q];
      Q[k * 4096 + idx] = out[q];
    }
    __syncthreads();
  }
}

// ---------------------------------------------------------------------------
// Triangular-Toeplitz kernel, CPB chunks per block (128 threads = 4 waves,
// wave w owns output n-tile w for all CPB chunks):
//   R_cc (16x64): forcing rows.  If Bmat!=null, R = Vtile @ Bmat^T (WMMA),
//                 else R = 16 consecutive rows of Vsrc.
//   Out_cc[j] = sum_{m<j} R_cc[m] @ P_{j-1-m},  j=0..15
//   Carry_cc  = Out_cc[15] @ P_1 + R_cc[15]
// P_k double-buffered in LDS via async B128 copies.
// ---------------------------------------------------------------------------
__global__ __launch_bounds__(128) void tri_scan_kernel(
    const float* __restrict__ Vsrc,   // tile rows source
    const float* __restrict__ Bmat,   // 64x64 raw [n][k], or nullptr
    const float* __restrict__ Pmat,   // 17 x 64 x 64 powers
    float* __restrict__ OutRows,      // (tiles*16) x 64
    float* __restrict__ CarryOut)     // tiles x 64
{
  __shared__ __align__(16) float Pd[2][NST][LDSP];  // async-staged powers
  __shared__ float Rlds[CPB][CH][LDSP];
  __shared__ float Bl[NST][LDSP];
  __shared__ float row15[CPB][NST];

  const int tile0 = blockIdx.x * CPB;
  const int tid  = threadIdx.x;
  const int lane = tid & 31;
  const int wave = tid >> 5;   // n-tile 0..3
  const int mrow = lane & 15;  // M row (A/D) and N col (B/D)
  const int half = lane >> 4;
  const int ncol = wave * 16 + mrow;

  // Kick off P_0 fill immediately; it overlaps all of the R-staging/GEMM.
  async_stage_pk(Pmat, &Pd[0][0][0], tid);

  if (Bmat != nullptr) {
    // Stage V tiles and raw B into LDS, compute R_cc = V_cc @ B^T via WMMA.
    for (int idx = tid; idx < CPB * CH * NST; idx += 128) {
      const int cc = idx >> 10;
      const int r = (idx >> 6) & 15, c = idx & 63;
      Rlds[cc][r][c] = Vsrc[tile0 * (CH * NST) + idx];
    }
    for (int idx = tid; idx < 4096; idx += 128)
      Bl[idx >> 6][idx & 63] = Bmat[idx];
    __syncthreads();

    v8f w[CPB];
#pragma unroll
    for (int cc = 0; cc < CPB; ++cc) {
      w[cc] = (v8f){0.f, 0.f, 0.f, 0.f, 0.f, 0.f, 0.f, 0.f};
      for (int kk = 0; kk < 16; ++kk) {
        const int K0 = 4 * kk + 2 * half;
        v2f a, b;
        a.x = Rlds[cc][mrow][K0];
        a.y = Rlds[cc][mrow][K0 + 1];
        b.x = Bl[ncol][K0];  // B'[K][N] = Bmat[N][K]
        b.y = Bl[ncol][K0 + 1];
        w[cc] = wmma_f32_16x16x4(a, b, w[cc]);
      }
    }
    __syncthreads();  // all B/V reads done before Rlds overwrite
#pragma unroll
    for (int cc = 0; cc < CPB; ++cc)
      for (int r = 0; r < 8; ++r) Rlds[cc][r + 8 * half][ncol] = w[cc][r];
    __syncthreads();
  } else {
    for (int idx = tid; idx < CPB * CH * NST; idx += 128) {
      const int cc = idx >> 10;
      const int r = (idx >> 6) & 15, c = idx & 63;
      Rlds[cc][r][c] = Vsrc[tile0 * (CH * NST) + idx];
    }
    __syncthreads();
  }

  // Out = sum_{k=0}^{14} ShiftDown_{k+1}(R) @ P_k, double-buffered P_k.
  v8f acc[CPB];
#pragma unroll
  for (int cc = 0; cc < CPB; ++cc)
    acc[cc] = (v8f){0.f, 0.f, 0.f, 0.f, 0.f, 0.f, 0.f, 0.f};

  for (int k = 0; k <= 14; ++k) {
    // Prefetch next power into the other buffer (read of that buffer last
    // finished at iteration k-1; the trailing barrier there makes it safe).
    if (k < 14) {
      async_stage_pk(Pmat + (k + 1) * 4096, &Pd[(k + 1) & 1][0][0], tid);
      asm volatile("s_wait_asynccnt 0x8" ::: "memory");  // cur 8 landed
    } else {
      asm volatile("s_wait_asynccnt 0x0" ::: "memory");
    }
    __syncthreads();

    const int src = mrow - 1 - k;
#pragma unroll
    for (int cc = 0; cc < CPB; ++cc) {
      for (int kk = 0; kk < 16; ++kk) {
        const int K0 = 4 * kk + 2 * half;
        v2f a, b;
        a.x = (src >= 0) ? Rlds[cc][src][K0] : 0.f;
        a.y = (src >= 0) ? Rlds[cc][src][K0 + 1] : 0.f;
        b.x = Pd[k & 1][K0][ncol];
        b.y = Pd[k & 1][K0 + 1][ncol];
        acc[cc] = wmma_f32_16x16x4(a, b, acc[cc]);
      }
    }
    __syncthreads();  // everyone done reading Pd[k&1] before overwrite
  }

  // D layout: VGPR r -> M = r + 8*half, N = ncol
#pragma unroll
  for (int cc = 0; cc < CPB; ++cc)
    for (int r = 0; r < 8; ++r) {
      const int M = r + 8 * half;
      OutRows[((tile0 + cc) * CH + M) * NST + ncol] = acc[cc][r];
      if (r == 7 && half == 1) row15[cc][ncol] = acc[cc][r];
    }
  __syncthreads();

  // Carry_cc = R_cc[15] + Out_cc[15] @ P_1   (two chunks per 64-thread group)
  {
    const int sub = tid >> 6, n = tid & 63;
    const float* P1 = Pmat + 4096;
    for (int pass = 0; pass < CPB / 2; ++pass) {
      const int cc = pass * 2 + sub;
      float s = Rlds[cc][15][n];
      for (int m = 0; m < NST; ++m) s += row15[cc][m] * P1[m * 64 + n];
      CarryOut[(tile0 + cc) * NST + n] = s;
    }
  }
}

// ---------------------------------------------------------------------------
// Serial scan over 32 group carries: g_{i+1} = g_i Q16 + GC[i]; stores starts.
// ---------------------------------------------------------------------------
__global__ __launch_bounds__(64) void group_scan_kernel(
    const float* __restrict__ GC, const float* __restrict__ Q16,
    float* __restrict__ Gstart)
{
  __shared__ float g[NST];
  const int n = threadIdx.x;
  g[n] = 0.f;
  __syncthreads();
  for (int gi = 0; gi < NG; ++gi) {
    Gstart[gi * NST + n] = g[n];
    float acc = GC[gi * NST + n];
    for (int m = 0; m < NST; ++m) acc += g[m] * Q16[m * 64 + n];
    __syncthreads();
    g[n] = acc;
    __syncthreads();
  }
}

// ---------------------------------------------------------------------------
// Chunk-start states: s_{16g+j} = Gstart[g] @ Q_j + Sloc[16g+j]
// ---------------------------------------------------------------------------
__global__ __launch_bounds__(64) void chunk_state_kernel(
    const float* __restrict__ Sloc, const float* __restrict__ Gstart,
    const float* __restrict__ Q, float* __restrict__ Schunk)
{
  __shared__ float gv[NST];
  const int c = blockIdx.x, n = threadIdx.x;
  const int g = c >> 4, j = c & 15;
  gv[n] = Gstart[g * NST + n];
  __syncthreads();
  const float* Qj = Q + j * 4096;
  float acc = Sloc[c * NST + n];
  for (int m = 0; m < NST; ++m) acc += gv[m] * Qj[m * 64 + n];
  Schunk[c * NST + n] = acc;
}

// ---------------------------------------------------------------------------
// Finalize: X[16c+j] += s_c @ P_j (store trajectory), then Y += X @ C_i^T.
// ---------------------------------------------------------------------------
__global__ __launch_bounds__(128) void finalize_kernel(
    float* __restrict__ Xtraj,        // in: local rows; out: final rows
    const float* __restrict__ Schunk, // NCH x 64
    const float* __restrict__ Pmat,   // powers
    const float* __restrict__ Cmat,   // C_i raw [y][k]
    float* __restrict__ Yout)         // T x 64, accumulated across layers
{
  __shared__ float Xt[CH][LDSP];
  __shared__ float Cl[NST][LDSP];
  __shared__ float sv[NST];

  const int c = blockIdx.x, tid = threadIdx.x;
  for (int idx = tid; idx < CH * NST; idx += 128)
    Xt[idx >> 6][idx & 63] = Xtraj[c * (CH * NST) + idx];
  if (tid < NST) sv[tid] = Schunk[c * NST + tid];
  for (int idx = tid; idx < 4096; idx += 128)
    Cl[idx >> 6][idx & 63] = Cmat[idx];
  __syncthreads();

  for (int q = 0; q < 8; ++q) {
    const int idx = tid + 128 * q;
    const int j = idx >> 6, n = idx & 63;
    const float* Pj = Pmat + j * 4096 + n;
    float acc = Xt[j][n];
    for (int m = 0; m < NST; ++m) acc += sv[m] * Pj[m * 64];
    Xt[j][n] = acc;
  }
  __syncthreads();
  for (int idx = tid; idx < CH * NST; idx += 128)
    Xtraj[c * (CH * NST) + idx] = Xt[idx >> 6][idx & 63];

  const int lane = tid & 31, wave = tid >> 5;
  const int mrow = lane & 15, half = lane >> 4;
  const int ncol = wave * 16 + mrow;
  v8f y = {0.f, 0.f, 0.f, 0.f, 0.f, 0.f, 0.f, 0.f};
  for (int kk = 0; kk < 16; ++kk) {
    const int K0 = 4 * kk + 2 * half;
    v2f a, b;
    a.x = Xt[mrow][K0];
    a.y = Xt[mrow][K0 + 1];
    b.x = Cl[ncol][K0];  // C'[K][Y] = Cmat[Y][K]
    b.y = Cl[ncol][K0 + 1];
    y = wmma_f32_16x16x4(a, b, y);
  }
  for (int r = 0; r < 8; ++r) {
    const int M = r + 8 * half;
    Yout[(c * CH + M) * NST + ncol] += y[r];  // layers serialized on stream
  }
}

__global__ void zero_kernel(float* __restrict__ p, int n) {
  const int i = blockIdx.x * 256 + threadIdx.x;
  if (i < n) p[i] = 0.f;
}

// ---------------------------------------------------------------------------
extern "C" void kernel_launch(void* const* d_in, const int* in_sizes, int n_in,
                              void* d_out, int out_size, void* d_ws,
                              size_t ws_size, hipStream_t stream)
{
  (void)in_sizes; (void)n_in; (void)out_size; (void)ws_size;

  const float* u    = (const float*)d_in[0];  // T x 64
  const float* Astk = (const float*)d_in[1];  // 32 x 64 x 64
  const float* B0   = (const float*)d_in[2];  // 64 x 64
  const float* Blow = (const float*)d_in[3];  // 31 x 64 x 64
  const float* Cstk = (const float*)d_in[4];  // 32 x 64 x 64
  // d_in[5] = D (zeros) -> ignored

  float* ws     = (float*)d_ws;
  float* Traj0  = ws;  ws += T_SEQ * NST;
  float* Traj1  = ws;  ws += T_SEQ * NST;
  float* CarryB = ws;  ws += NCH * NST;
  float* SlocB  = ws;  ws += NCH * NST;
  float* GCB    = ws;  ws += NG * NST;
  float* Gst    = ws;  ws += NG * NST;
  float* Schnk  = ws;  ws += NCH * NST;
  float* P      = ws;  ws += 17 * 4096;
  float* Q      = ws;  ws += 17 * 4096;
  float* Y      = (float*)d_out;

  zero_kernel<<<(T_SEQ * NST + 255) / 256, 256, 0, stream>>>(Y, T_SEQ * NST);

  for (int i = 0; i < NLAY; ++i) {
    const float* Ai = Astk + i * 4096;
    const float* Bi = (i == 0) ? B0 : (Blow + (i - 1) * 4096);
    const float* Vi = (i == 0) ? u : ((i & 1) ? Traj0 : Traj1);
    float*       Xi = (i & 1) ? Traj1 : Traj0;
    const float* Ci = Cstk + i * 4096;

    powers_kernel<<<1, 256, 0, stream>>>(Ai, P, Q);
    // chunk-local solutions + chunk carries (WMMA + async double-buffer)
    tri_scan_kernel<<<NCH / CPB, 128, 0, stream>>>(Vi, Bi, P, Xi, CarryB);
    // group-local prefix over carries (same kernel, Q powers)
    tri_scan_kernel<<<NG / CPB, 128, 0, stream>>>(CarryB, nullptr, Q, SlocB,
                                                  GCB);
    // 32-step serial spine
    group_scan_kernel<<<1, 64, 0, stream>>>(GCB, Q + 16 * 4096, Gst);
    chunk_state_kernel<<<NCH, 64, 0, stream>>>(SlocB, Gst, Q, Schnk);
    // apply chunk-start states, emit trajectory + fused Y projection (WMMA)
    finalize_kernel<<<NCH, 128, 0, stream>>>(Xi, Schnk, P, Ci, Y);
  }
}